// PointCloudGenerator_33646773796929
// MI455X (gfx1250) — compile-verified
//
#include <hip/hip_runtime.h>
#include <hip/hip_bf16.h>

// ---------------- problem constants ----------------
#define BB   8
#define CC   62
#define GG   26
#define G3   (GG*GG*GG)     // 17576
#define NS   32768
#define HH   1024
#define CIN  64             // C + 2
#define ST   2              // sample tiles (of 16) per wave

typedef float v2f __attribute__((ext_vector_type(2)));
typedef float v8f __attribute__((ext_vector_type(8)));

// ---------------- kernel 1: per-batch inclusive scan of dens, zero hist ----------------
__global__ void __launch_bounds__(512)
k_scan_dens(const float* __restrict__ dens, float* __restrict__ csum,
            int* __restrict__ hist) {
    const int b = blockIdx.x;
    const int t = threadIdx.x;
    __shared__ float tile[512];
    float carry = 0.0f;
    for (int base = 0; base < G3; base += 512) {
        const int i = base + t;
        float v = (i < G3) ? dens[b * G3 + i] : 0.0f;
        tile[t] = v;
        __syncthreads();
        #pragma unroll
        for (int off = 1; off < 512; off <<= 1) {
            float x = (t >= off) ? tile[t - off] : 0.0f;
            __syncthreads();
            tile[t] += x;
            __syncthreads();
        }
        if (i < G3) csum[b * G3 + i] = tile[t] + carry;
        __syncthreads();
        carry += tile[511];          // uniform across threads
        __syncthreads();
    }
    for (int i = t; i < G3; i += 512) hist[b * G3 + i] = 0;
}

// ---------------- kernel 2: searchsorted(right) + histogram ----------------
__global__ void __launch_bounds__(256)
k_sample(const float* __restrict__ csum, const float* __restrict__ u,
         int* __restrict__ hist) {
    const int n = blockIdx.x * 256 + threadIdx.x;
    const int b = blockIdx.y;
    const float* cs = csum + (size_t)b * G3;
    const float total  = cs[G3 - 1];
    const float target = u[(size_t)b * NS + n] * total;
    // upper_bound: first position where cs[pos] > target
    int lo = 0, hi = G3;
    while (lo < hi) {
        int mid = (lo + hi) >> 1;
        if (cs[mid] <= target) lo = mid + 1; else hi = mid;
    }
    int idx = (lo < G3) ? lo : (G3 - 1);      // clip to [0, G3-1]
    atomicAdd(&hist[(size_t)b * G3 + idx], 1);
}

// ---------------- kernel 3: per-batch exclusive scan of hist ----------------
__global__ void __launch_bounds__(512)
k_scan_hist(const int* __restrict__ hist, int* __restrict__ offs) {
    const int b = blockIdx.x;
    const int t = threadIdx.x;
    __shared__ int tile[512];
    int carry = 0;
    for (int base = 0; base < G3; base += 512) {
        const int i = base + t;
        int v = (i < G3) ? hist[b * G3 + i] : 0;
        tile[t] = v;
        __syncthreads();
        #pragma unroll
        for (int off = 1; off < 512; off <<= 1) {
            int x = (t >= off) ? tile[t - off] : 0;
            __syncthreads();
            tile[t] += x;
            __syncthreads();
        }
        if (i < G3) offs[b * G3 + i] = tile[t] - v + carry;   // exclusive
        __syncthreads();
        carry += tile[511];
        __syncthreads();
    }
}

// ---------------- kernel 4: expand counting sort -> sorted indices ----------------
__global__ void __launch_bounds__(256)
k_expand(const int* __restrict__ hist, const int* __restrict__ offs,
         int* __restrict__ sidx) {
    const int k = blockIdx.x * 256 + threadIdx.x;
    const int b = blockIdx.y;
    if (k >= G3) return;
    const int cnt = hist[(size_t)b * G3 + k];
    const int off = offs[(size_t)b * G3 + k];
    for (int j = 0; j < cnt; ++j) sidx[(size_t)b * NS + off + j] = k;
}

// ---------------- kernel 5: fused MLP via V_WMMA_F32_16X16X4_F32 ----------------
// Workgroup = 8 waves, 256 samples. Per H-row tile r, the 16x64 W1 tile (4 KB)
// is staged to LDS once and reused by all 8 waves x 2 sample tiles (16x traffic cut).
// WMMA layouts (wave32):
//   A (16x4 f32):  lanes 0-15 -> M=lane, K=0..1 in v[0..1]; lanes 16-31 -> K=2..3
//   B (4x16 f32):  lanes 0-15 -> N=lane, K=0..1 in v[0..1]; lanes 16-31 -> K=2..3
//   C/D (16x16):   VGPR j: lanes 0-15 -> M=j, lanes 16-31 -> M=j+8; N=lane%16
__global__ void __launch_bounds__(256)
k_mlp(const float* __restrict__ x, const float* __restrict__ rnd,
      const int* __restrict__ sidx,
      const float* __restrict__ W1, const float* __restrict__ b1,
      const float* __restrict__ W2, const float* __restrict__ b2,
      float* __restrict__ out) {
    __shared__ float w1s[16 * CIN];   // 4 KB: current A tile (16 rows x 64 cols)
    __shared__ float b1s[HH];         // 4 KB
    __shared__ float w2s[3 * HH];     // 12 KB

    const int tid  = threadIdx.x;
    const int b    = blockIdx.y;
    const int wid  = tid >> 5;
    const int lane = tid & 31;
    const int half = lane >> 4;          // 0: K%4 in {0,1}, 1: K%4 in {2,3}
    const int col  = lane & 15;          // sample column within tile

    // Stage b1 and W2 once per workgroup.
    #pragma unroll
    for (int i = tid; i < HH; i += 256)     b1s[i] = b1[i];
    #pragma unroll
    for (int i = tid; i < 3 * HH; i += 256) w2s[i] = W2[i];

    const float* xb = x + (size_t)b * CC * G3;

    // Stage binp (64 x 16) tiles into registers in WMMA B layout.
    int nn[ST], si[ST];
    v2f bt[ST][16];
    #pragma unroll
    for (int t = 0; t < ST; ++t) {
        nn[t] = blockIdx.x * (256) + (wid * ST + t) * 16 + col;
        si[t] = sidx[(size_t)b * NS + nn[t]];
        #pragma unroll
        for (int s = 0; s < 16; ++s) {
            const int c0 = 4 * s + 2 * half;
            const int c1 = c0 + 1;
            float v0 = (c0 < CC) ? xb[(size_t)c0 * G3 + si[t]]
                                 : rnd[((size_t)b * 2 + (c0 - CC)) * NS + nn[t]] * 2.0f - 1.0f;
            float v1 = (c1 < CC) ? xb[(size_t)c1 * G3 + si[t]]
                                 : rnd[((size_t)b * 2 + (c1 - CC)) * NS + nn[t]] * 2.0f - 1.0f;
            bt[t][s].x = v0;
            bt[t][s].y = v1;
        }
    }

    float o0[ST], o1[ST], o2[ST];
    #pragma unroll
    for (int t = 0; t < ST; ++t) { o0[t] = 0.0f; o1[t] = 0.0f; o2[t] = 0.0f; }

    #pragma unroll 1
    for (int r = 0; r < HH / 16; ++r) {
        // Cooperative stage of the 16x64 W1 tile into LDS (4 floats/thread).
        __syncthreads();   // previous iteration's LDS reads complete
        ((float4*)w1s)[tid] = ((const float4*)(W1 + (size_t)r * 16 * CIN))[tid];
        __syncthreads();

        // Per-lane A registers from LDS (shared across both sample tiles).
        v2f a[16];
        const v2f* wp = (const v2f*)(w1s + col * CIN + 2 * half);
        #pragma unroll
        for (int s = 0; s < 16; ++s) a[s] = wp[2 * s];

        #pragma unroll
        for (int t = 0; t < ST; ++t) {
            // C init = b1 broadcast in C layout
            v8f acc;
            #pragma unroll
            for (int j = 0; j < 8; ++j) acc[j] = b1s[r * 16 + j + 8 * half];

            #pragma unroll
            for (int s = 0; s < 16; ++s) {
                acc = __builtin_amdgcn_wmma_f32_16x16x4_f32(
                    /*neg_a=*/false, a[s], /*neg_b=*/false, bt[t][s],
                    /*c_mod=*/(short)0, acc, /*reuse_a=*/false, /*reuse_b=*/false);
            }

            // ReLU + W2 partial reduction (each lane covers 8 of the 16 rows)
            #pragma unroll
            for (int j = 0; j < 8; ++j) {
                float h = acc[j] > 0.0f ? acc[j] : 0.0f;
                const int hidx = r * 16 + j + 8 * half;
                o0[t] = fmaf(w2s[0 * HH + hidx], h, o0[t]);
                o1[t] = fmaf(w2s[1 * HH + hidx], h, o1[t]);
                o2[t] = fmaf(w2s[2 * HH + hidx], h, o2[t]);
            }
        }
    }

    // Epilogue: combine half-wave partials, bias, norm/reg, grid offset, store.
    #pragma unroll
    for (int t = 0; t < ST; ++t) {
        float a0 = o0[t] + __shfl_xor(o0[t], 16);
        float a1 = o1[t] + __shfl_xor(o1[t], 16);
        float a2 = o2[t] + __shfl_xor(o2[t], 16);
        if (half == 0) {
            a0 += b2[0]; a1 += b2[1]; a2 += b2[2];
            const float nrm = sqrtf(a0 * a0 + a1 * a1 + a2 * a2);
            const float thr = 1.7320508075688772f / (float)GG;   // sqrt(3)/G
            const float reg = fmaxf(nrm - thr, 0.0f);

            const int s  = si[t];
            const int ix = s / (GG * GG);
            const int rm = s - ix * (GG * GG);
            const int iy = rm / GG;
            const int iz = rm - iy * GG;
            const float sc = 2.0f / (float)GG;
            const float ox = ((float)ix + 0.5f) * sc - 1.0f;
            const float oy = ((float)iy + 0.5f) * sc - 1.0f;
            const float oz = ((float)iz + 0.5f) * sc - 1.0f;

            const int n = nn[t];
            out[((size_t)b * 3 + 0) * NS + n] = a0 + ox;
            out[((size_t)b * 3 + 1) * NS + n] = a1 + oy;
            out[((size_t)b * 3 + 2) * NS + n] = a2 + oz;
            out[(size_t)BB * 3 * NS + (size_t)b * NS + n] = reg;
        }
    }
}

// ---------------- host launcher ----------------
extern "C" void kernel_launch(void* const* d_in, const int* in_sizes, int n_in,
                              void* d_out, int out_size, void* d_ws, size_t ws_size,
                              hipStream_t stream) {
    (void)in_sizes; (void)n_in; (void)out_size; (void)ws_size;

    const float* x    = (const float*)d_in[0];
    const float* dens = (const float*)d_in[1];
    const float* rnd  = (const float*)d_in[2];
    const float* u    = (const float*)d_in[3];
    const float* W1   = (const float*)d_in[4];
    const float* b1   = (const float*)d_in[5];
    const float* W2   = (const float*)d_in[6];
    const float* b2   = (const float*)d_in[7];
    float* out = (float*)d_out;

    // workspace carve-up
    float* csum = (float*)d_ws;                       // B*G3 floats
    int*   hist = (int*)(csum + (size_t)BB * G3);     // B*G3 ints
    int*   offs = hist + (size_t)BB * G3;             // B*G3 ints
    int*   sidx = offs + (size_t)BB * G3;             // B*NS ints

    k_scan_dens<<<dim3(BB), dim3(512), 0, stream>>>(dens, csum, hist);
    k_sample  <<<dim3(NS / 256, BB), dim3(256), 0, stream>>>(csum, u, hist);
    k_scan_hist<<<dim3(BB), dim3(512), 0, stream>>>(hist, offs);
    k_expand  <<<dim3((G3 + 255) / 256, BB), dim3(256), 0, stream>>>(hist, offs, sidx);
    k_mlp     <<<dim3(NS / 256, BB), dim3(256), 0, stream>>>(x, rnd, sidx, W1, b1, W2, b2, out);
}